// HomomorphicVocoder_29686813950825
// MI455X (gfx1250) — compile-verified
//
#include <hip/hip_runtime.h>
#include <hip/hip_bf16.h>

// ---------------------------------------------------------------------------
// HomomorphicVocoder on MI455X (gfx1250, wave32)
//  B=16, T=600, HOP=256, N_FFT=1024, F=513, frames=601, SR=24000
//  - mel->linear projections via V_WMMA_F32_16X16X4_F32 (fp32 WMMA),
//    branch-free operand fetch, K fully unrolled (templated D)
//  - 1024-pt Stockham radix-2 FFTs in LDS (stft / min-phase cepstrum / istft)
//  - gather-style overlap-add (deterministic, no atomics)
// ---------------------------------------------------------------------------

#define Bn     16
#define Tn     600
#define HOPn   256
#define NFFT   1024
#define NFREQ  513
#define NFRM   601
#define LSIG   (Tn * HOPn)        // 153600
#define SRf    24000.0f
#define D_PER  16
#define D_ENV  80

typedef float v8f __attribute__((ext_vector_type(8)));
typedef float v2f __attribute__((ext_vector_type(2)));

__device__ __forceinline__ float hannw(int j) {
    // periodic hann: 0.5 - 0.5*cos(2*pi*j/1024)
    return 0.5f - 0.5f * __cosf((float)j * 6.1359231515425649e-3f);
}

__device__ __forceinline__ float2 cmulf(float2 a, float2 b) {
    return make_float2(a.x * b.x - a.y * b.y, a.x * b.y + a.y * b.x);
}

// ---------------------------------------------------------------------------
// 1024-point complex Stockham radix-2 DIF FFT in LDS (self-sorting).
// sign = -1 forward, +1 inverse (caller applies 1/N for inverse).
// Data starts and ends in b0 (10 stages -> even # of ping-pongs).
// ---------------------------------------------------------------------------
__device__ void fft1024_lds(float2* b0, float2* b1, int tid, int nthr, float sign) {
    float2* src = b0;
    float2* dst = b1;
    int ls = 0; // log2(s)
    for (int stage = 0; stage < 10; ++stage) {
        const int s = 1 << ls;
        const int m = 512 >> ls;           // half of current transform length
        __syncthreads();
        for (int i = tid; i < 512; i += nthr) {
            const int q = i & (s - 1);
            const int p = i >> ls;
            float2 u = src[q + s * p];
            float2 v = src[q + s * (p + m)];
            float2 su = make_float2(u.x + v.x, u.y + v.y);
            float2 di = make_float2(u.x - v.x, u.y - v.y);
            float ang = sign * 3.14159265358979323846f * (float)p / (float)m;
            float sn, cs;
            __sincosf(ang, &sn, &cs);
            dst[q + s * (2 * p)]     = su;
            dst[q + s * (2 * p + 1)] = cmulf(di, make_float2(cs, sn));
        }
        float2* t = src; src = dst; dst = t;
        ++ls;
    }
    __syncthreads();  // result resides in b0
}

// ---------------------------------------------------------------------------
// Compute mel filterbank F [513 x nm] and INV = F * (F^T F)^-1  (== pinv(F^T))
// One workgroup; Gauss-Jordan on the Gram matrix in LDS.
// ---------------------------------------------------------------------------
#define MAXM 80
__global__ void fbinv_kernel(int nm, float* __restrict__ Fg, float* __restrict__ INV) {
    __shared__ float sG[MAXM * MAXM];
    __shared__ float sI[MAXM * MAXM];
    __shared__ float sFac[MAXM];
    __shared__ float fpts[MAXM + 2];
    const int tid = threadIdx.x, nthr = blockDim.x;

    const float mmax = 2595.0f * log10f(1.0f + 12000.0f / 700.0f);
    for (int j = tid; j < nm + 2; j += nthr) {
        float m = mmax * (float)j / (float)(nm + 1);
        fpts[j] = 700.0f * (powf(10.0f, m / 2595.0f) - 1.0f);
    }
    __syncthreads();

    // triangular filterbank
    for (int i = tid; i < NFREQ * nm; i += nthr) {
        int k = i / nm, j = i % nm;
        float fr = (float)k * (12000.0f / 512.0f);
        float dn = (fr - fpts[j]) / (fpts[j + 1] - fpts[j]);
        float up = (fpts[j + 2] - fr) / (fpts[j + 2] - fpts[j + 1]);
        Fg[i] = fmaxf(0.0f, fminf(dn, up));
    }
    __syncthreads();

    // Gram matrix G = F^T F, identity alongside
    for (int i = tid; i < nm * nm; i += nthr) {
        int r = i / nm, c = i % nm;
        float acc = 0.0f;
        for (int k = 0; k < NFREQ; ++k) acc += Fg[k * nm + r] * Fg[k * nm + c];
        sG[i] = acc;
        sI[i] = (r == c) ? 1.0f : 0.0f;
    }
    __syncthreads();

    // Gauss-Jordan (SPD: no pivoting needed)
    for (int col = 0; col < nm; ++col) {
        float pinv = 1.0f / sG[col * nm + col];
        __syncthreads();
        for (int j = tid; j < nm; j += nthr) {
            sG[col * nm + j] *= pinv;
            sI[col * nm + j] *= pinv;
        }
        __syncthreads();
        for (int r = tid; r < nm; r += nthr)
            sFac[r] = (r == col) ? 0.0f : sG[r * nm + col];
        __syncthreads();
        for (int i = tid; i < nm * nm; i += nthr) {
            int r = i / nm, j = i % nm;
            float f = sFac[r];
            sG[i] -= f * sG[col * nm + j];
            sI[i] -= f * sI[col * nm + j];
        }
        __syncthreads();
    }

    // INV = F * Ginv   [513 x nm]
    for (int i = tid; i < NFREQ * nm; i += nthr) {
        int k = i / nm, j = i % nm;
        float acc = 0.0f;
        for (int q = 0; q < nm; ++q) acc += Fg[k * nm + q] * sI[q * nm + j];
        INV[i] = acc;
    }
}

// ---------------------------------------------------------------------------
// Linear upsample of f0 by HOP (align_corners=False semantics)
// ---------------------------------------------------------------------------
__device__ __forceinline__ float f0_upsample(const float* __restrict__ f0row, int n) {
    float pos = ((float)n + 0.5f) * (1.0f / (float)HOPn) - 0.5f;
    pos = fminf(fmaxf(pos, 0.0f), (float)(Tn - 1));
    int lo = (int)floorf(pos);
    int hi = min(lo + 1, Tn - 1);
    float w = pos - (float)lo;
    return f0row[lo] * (1.0f - w) + f0row[hi] * w;
}

// Per-batch sequential-carry block scan: saw[b,n] = (cumsum(f0_up)/SR) mod 1
__global__ void saw_kernel(const float* __restrict__ f0, float* __restrict__ saw) {
    const int b = blockIdx.x;
    const int tid = threadIdx.x;
    __shared__ float sb[256];
    const float* f0row = f0 + b * Tn;
    float carry = 0.0f;
    for (int c = 0; c < LSIG; c += 256) {
        int n = c + tid;
        float v = f0_upsample(f0row, n);
        __syncthreads();            // protect prior-chunk sb[255] read
        sb[tid] = v;
        __syncthreads();
        for (int off = 1; off < 256; off <<= 1) {
            float add = (tid >= off) ? sb[tid - off] : 0.0f;
            __syncthreads();
            sb[tid] += add;
            __syncthreads();
        }
        float tot = carry + sb[tid];
        saw[(size_t)b * LSIG + n] = fmodf(tot * (1.0f / SRf), 1.0f);
        carry += sb[255];
    }
}

// imp = (saw - roll(saw,-1) + f0_up/SR) * rsqrt(max(f0_up,20)) * sqrt(SR)
__global__ void imp_kernel(const float* __restrict__ f0, const float* __restrict__ saw,
                           float* __restrict__ imp) {
    int idx = blockIdx.x * blockDim.x + threadIdx.x;
    if (idx >= Bn * LSIG) return;
    int b = idx / LSIG, n = idx - b * LSIG;
    float fu = f0_upsample(f0 + b * Tn, n);
    float s0 = saw[idx];
    int n1 = (n + 1 == LSIG) ? 0 : n + 1;
    float s1 = saw[(size_t)b * LSIG + n1];
    float val = s0 - s1 + fu * (1.0f / SRf);
    float scale = rsqrtf(fmaxf(fu, 20.0f)) * 154.91933384829668f; // sqrt(24000)
    imp[idx] = val * scale;
}

// ---------------------------------------------------------------------------
// mel->linear projection via fp32 WMMA 16x16x4 (K fully unrolled).
//   out[b,t,f] = relu( sum_k INV[f,k] * mel_pad[b,k,t] ) (* voicing mask)
// One wave per 16x16 (f,t) tile.
// A fragment: lane<16 holds A[M=lane][K=k0,k0+1]; lane>=16 holds K=k0+2,k0+3.
// B fragment mirrors A; C/D: vgpr r = row (m = r + 8*half), lane%16 = col.
// Out-of-range handled branch-free: clamp index, zero via 0/1 multiplier.
// ---------------------------------------------------------------------------
template <int D, int USEMASK>
__global__ void __launch_bounds__(32)
mel2lin_wmma(const float* __restrict__ inv,   // [513, D]
             const float* __restrict__ mel,   // [B, D, 600]
             const float* __restrict__ f0,    // [B, 600] (mask) or nullptr
             float* __restrict__ out)         // [B, 601, 513]
{
    const int lane  = threadIdx.x;
    const int half  = lane >> 4;
    const int r     = lane & 15;
    const int fbase = blockIdx.x * 16;
    const int tbase = blockIdx.y * 16;
    const int b     = blockIdx.z;

    v8f acc = {};

    const int   f_a  = fbase + r;                      // A row for this lane
    const int   f_c  = min(f_a, NFREQ - 1);            // clamped (always valid)
    const float fsel = (f_a < NFREQ) ? 1.0f : 0.0f;
    const int   t_b  = tbase + r;                      // padded-frame column
    const bool  tok  = (t_b >= 1) && (t_b <= Tn);      // frame 0 is zero pad
    const float tsel = tok ? 1.0f : 0.0f;
    const int   t_c  = min(max(t_b - 1, 0), Tn - 1);   // clamped mel column

    const float2* __restrict__ ap = (const float2*)(inv + f_c * D);  // 8B aligned
    const float*  __restrict__ bp = mel + (size_t)b * D * Tn + t_c;

    #pragma unroll
    for (int kt = 0; kt < D; kt += 4) {
        const int ka = kt + half * 2;                  // even
        float2 av = ap[ka >> 1];
        v2f a, bb;
        a.x  = av.x * fsel;
        a.y  = av.y * fsel;
        bb.x = bp[(size_t)ka * Tn] * tsel;
        bb.y = bp[(size_t)(ka + 1) * Tn] * tsel;
        acc = __builtin_amdgcn_wmma_f32_16x16x4_f32(
            false, a, false, bb, (short)0, acc, false, false);
    }

    float mask = 1.0f;
    if (USEMASK) {
        float fv = f0[b * Tn + t_c];
        mask = (tok && fv > 20.0f) ? 1.0f : 0.0f;
    }
    if (t_b < NFRM) {
        size_t obase = ((size_t)b * NFRM + t_b) * NFREQ;
        #pragma unroll
        for (int v = 0; v < 8; ++v) {
            int f = fbase + v + half * 8;
            if (f < NFREQ) out[obase + f] = fmaxf(acc[v], 0.0f) * mask;
        }
    }
}

// ---------------------------------------------------------------------------
// STFT: one workgroup per (frame, batch, signal). Reflect pad, window, FFT.
// Output layout [B, 601, 513] float2 (frame-major for downstream LDS kernels).
// ---------------------------------------------------------------------------
__global__ void stft_kernel(const float* __restrict__ sigA, const float* __restrict__ sigB,
                            float2* __restrict__ outA, float2* __restrict__ outB) {
    __shared__ float2 X[NFFT];
    __shared__ float2 Y[NFFT];
    const int t = blockIdx.x, b = blockIdx.y, which = blockIdx.z;
    const int tid = threadIdx.x, nthr = blockDim.x;
    const float* sig = which ? sigB : sigA;
    float2* out = which ? outB : outA;

    for (int j = tid; j < NFFT; j += nthr) {
        int k0 = t * HOPn + j - (NFFT / 2);
        if (k0 < 0) k0 = -k0;
        else if (k0 >= LSIG) k0 = 2 * LSIG - 2 - k0;
        X[j] = make_float2(sig[(size_t)b * LSIG + k0] * hannw(j), 0.0f);
    }
    fft1024_lds(X, Y, tid, nthr, -1.0f);
    size_t obase = ((size_t)b * NFRM + t) * NFREQ;
    for (int f = tid; f < NFREQ; f += nthr) out[obase + f] = X[f];
}

// ---------------------------------------------------------------------------
// Minimum-phase reconstruction + spectral mixing, fused per (frame,batch):
//   amp = per*env ; mph = exp(fft(cwin * ifft(log amp_full)))[:513]
//   voi = imp_stft * mph + noi_stft * (1-per)*env   (written over imp_stft)
// ---------------------------------------------------------------------------
__global__ void minphase_mix_kernel(const float* __restrict__ per, const float* __restrict__ env,
                                    float2* __restrict__ impS /* in/out -> voi */,
                                    const float2* __restrict__ noiS) {
    __shared__ float2 X[NFFT];
    __shared__ float2 Y[NFFT];
    __shared__ float sPer[NFREQ];
    __shared__ float sEnv[NFREQ];
    const int t = blockIdx.x, b = blockIdx.y;
    const int tid = threadIdx.x, nthr = blockDim.x;
    const size_t base = ((size_t)b * NFRM + t) * NFREQ;

    for (int f = tid; f < NFREQ; f += nthr) {
        float p = per[base + f], e = env[base + f];
        sPer[f] = p; sEnv[f] = e;
        X[f] = make_float2(__logf(fmaxf(p * e, 1e-5f)), 0.0f);
    }
    __syncthreads();
    for (int f = NFREQ + tid; f < NFFT; f += nthr)
        X[f] = make_float2(X[NFFT - f].x, 0.0f);          // even symmetry

    fft1024_lds(X, Y, tid, nthr, +1.0f);                  // ifft (unscaled)

    for (int n = tid; n < NFFT; n += nthr) {              // cepstral window / N
        float w = (n == 0 || n == NFFT / 2) ? 1.0f : (n < NFFT / 2 ? 2.0f : 0.0f);
        float sc = w * (1.0f / (float)NFFT);
        X[n] = make_float2(X[n].x * sc, X[n].y * sc);
    }

    fft1024_lds(X, Y, tid, nthr, -1.0f);                  // fft

    for (int f = tid; f < NFREQ; f += nthr) {
        float2 c = X[f];
        float ex = __expf(c.x);
        float sn, cs; __sincosf(c.y, &sn, &cs);
        float2 mph = make_float2(ex * cs, ex * sn);
        float2 is = impS[base + f];
        float2 ns = noiS[base + f];
        float g = (1.0f - sPer[f]) * sEnv[f];
        impS[base + f] = make_float2(is.x * mph.x - is.y * mph.y + ns.x * g,
                                     is.x * mph.y + is.y * mph.x + ns.y * g);
    }
}

// ---------------------------------------------------------------------------
// iSTFT frame synthesis: hermitian-extend, inverse FFT, window, store frame.
// ---------------------------------------------------------------------------
__global__ void istft_frames_kernel(const float2* __restrict__ voi, float* __restrict__ frames) {
    __shared__ float2 X[NFFT];
    __shared__ float2 Y[NFFT];
    const int t = blockIdx.x, b = blockIdx.y;
    const int tid = threadIdx.x, nthr = blockDim.x;
    const size_t base = ((size_t)b * NFRM + t) * NFREQ;

    for (int f = tid; f < NFREQ; f += nthr) X[f] = voi[base + f];
    __syncthreads();
    for (int f = NFREQ + tid; f < NFFT; f += nthr) {
        float2 c = X[NFFT - f];
        X[f] = make_float2(c.x, -c.y);
    }
    fft1024_lds(X, Y, tid, nthr, +1.0f);                  // inverse (unscaled)
    size_t fb = ((size_t)b * NFRM + t) * NFFT;
    for (int j = tid; j < NFFT; j += nthr)
        frames[fb + j] = X[j].x * (1.0f / (float)NFFT) * hannw(j);
}

// ---------------------------------------------------------------------------
// Gather-style overlap-add + window-square normalization + center crop.
// ---------------------------------------------------------------------------
__global__ void ola_kernel(const float* __restrict__ frames, float* __restrict__ out) {
    int idx = blockIdx.x * blockDim.x + threadIdx.x;
    if (idx >= Bn * LSIG) return;
    int b = idx / LSIG, no = idx - b * LSIG;
    int n = no + NFFT / 2;
    int t_lo = (n >= NFFT - 1) ? ((n - (NFFT - 1) + HOPn - 1) >> 8) : 0;
    int t_hi = min(Tn, n >> 8);
    // prefetch first contributing frame line (global_prefetch_b8)
    __builtin_prefetch(&frames[((size_t)b * NFRM + t_lo) * NFFT + (n - (t_lo << 8))], 0, 1);
    float y = 0.0f, wsq = 0.0f;
    for (int t = t_lo; t <= t_hi; ++t) {
        int j = n - (t << 8);
        float w = hannw(j);
        y += frames[((size_t)b * NFRM + t) * NFFT + j];
        wsq += w * w;
    }
    out[idx] = y / fmaxf(wsq, 1e-11f);
}

// ---------------------------------------------------------------------------
extern "C" void kernel_launch(void* const* d_in, const int* in_sizes, int n_in,
                              void* d_out, int out_size, void* d_ws, size_t ws_size,
                              hipStream_t stream) {
    const float* f0     = (const float*)d_in[0];  // [16, 600]
    const float* per_in = (const float*)d_in[1];  // [16, 16, 600]
    const float* env_in = (const float*)d_in[2];  // [16, 80, 600]
    const float* noi    = (const float*)d_in[3];  // [16, 153600]
    float* out = (float*)d_out;                   // [16, 153600]

    float* ws = (float*)d_ws;
    size_t o = 0;
    float*  inv_per  = ws + o; o += (size_t)NFREQ * D_PER;           // 8.2K
    float*  inv_env  = ws + o; o += (size_t)NFREQ * D_ENV;           // 41K
    float*  fscratch = ws + o; o += (size_t)NFREQ * D_ENV;           // 41K
    float*  saw      = ws + o; o += (size_t)Bn * LSIG;               // 2.46M
    float*  impsig   = ws + o; o += (size_t)Bn * LSIG;               // 2.46M
    float*  perL     = ws + o; o += (size_t)Bn * NFRM * NFREQ;       // 4.93M
    float*  envL     = ws + o; o += (size_t)Bn * NFRM * NFREQ;       // 4.93M
    float2* impS     = (float2*)(ws + o); o += (size_t)Bn * NFRM * NFREQ * 2;
    float2* noiS     = (float2*)(ws + o); o += (size_t)Bn * NFRM * NFREQ * 2;
    float*  frames   = ws + o; o += (size_t)Bn * NFRM * NFFT;        // 9.85M

    // constants: pseudo-inverse mel projections (deterministic each call)
    fbinv_kernel<<<1, 256, 0, stream>>>(D_PER, fscratch, inv_per);
    fbinv_kernel<<<1, 256, 0, stream>>>(D_ENV, fscratch, inv_env);

    // impulse train
    saw_kernel<<<Bn, 256, 0, stream>>>(f0, saw);
    imp_kernel<<<(Bn * LSIG + 255) / 256, 256, 0, stream>>>(f0, saw, impsig);

    // mel -> linear spectra (WMMA), fused relu + voicing mask
    dim3 gmm((NFREQ + 15) / 16, (NFRM + 15) / 16, Bn);   // 33 x 38 x 16
    mel2lin_wmma<D_PER, 1><<<gmm, 32, 0, stream>>>(inv_per, per_in, f0, perL);
    mel2lin_wmma<D_ENV, 0><<<gmm, 32, 0, stream>>>(inv_env, env_in, nullptr, envL);

    // STFTs of impulse train and noise
    dim3 gs(NFRM, Bn, 2);
    stft_kernel<<<gs, 128, 0, stream>>>(impsig, noi, impS, noiS);

    // minimum phase + mix (voi written over impS)
    dim3 gf(NFRM, Bn);
    minphase_mix_kernel<<<gf, 128, 0, stream>>>(perL, envL, impS, noiS);

    // iSTFT frames then overlap-add
    istft_frames_kernel<<<gf, 128, 0, stream>>>(impS, frames);
    ola_kernel<<<(Bn * LSIG + 255) / 256, 256, 0, stream>>>(frames, out);
}